// XPooling_67413806678496
// MI455X (gfx1250) — compile-verified
//
#include <hip/hip_runtime.h>
#include <hip/hip_bf16.h>

#define NG      512        // graphs
#define NPG     195        // nodes per graph
#define NNODES  99840      // NG*NPG
#define CFEAT   128
#define NEDGE   1000000
#define SURV    193        // NPG - 2
#define NBNODES 99328      // NNODES - NG
#define CAP     4096       // per-graph edge-key capacity (mean 1953, sigma ~44)

typedef float v2f __attribute__((ext_vector_type(2)));
typedef float v8f __attribute__((ext_vector_type(8)));
typedef unsigned long long u64;

// ---------------- K0: init outputs / tables ----------------
__global__ void k_init(int* out_ei, int* out_batch, u64* table, int* cursor) {
  int stride = gridDim.x * blockDim.x;
  for (int i = blockIdx.x * blockDim.x + threadIdx.x; i < 2 * NEDGE; i += stride) {
    out_ei[i] = -1;
    if (i < NBNODES) out_batch[i] = (i < NG) ? i : (i - NG) / SURV;
    if (i < NG) { table[i] = 0ull; cursor[i] = 0; }
  }
}

// ---------------- K1: node projections p0 = x*W[:C], p1 = x*W[C:] via V_WMMA_F32_16X16X4_F32 ----
// A (16x4 f32): lanes 0-15 hold row M=lane K={0,1}; lanes 16-31 hold K={2,3}.
// B (4x16):     vgpr j, lanes 0-15 -> row K=j; lanes 16-31 -> row K=j+2; col N=lane%16.
// D (16x16):    vgpr i, lanes 0-15 -> M=i, N=lane; lanes 16-31 -> M=i+8.
// B fragment is built branch-free: per-lane base pointer chosen once + 0/1 mask multiply,
// so the unrolled inner loop is load/load/mul/mul/wmma with no exec-mask churn.
__global__ void k_project(const float* __restrict__ x, const float* __restrict__ W,
                          float* __restrict__ p0, float* __restrict__ p1) {
  int wave = (blockIdx.x * blockDim.x + threadIdx.x) >> 5;
  int lane = threadIdx.x & 31;
  if (wave >= NNODES / 16) return;            // uniform per wave (exact division: 6240 tiles)
  int m0   = wave * 16;
  int khi  = lane >> 4;                        // 0 or 1
  int n    = lane & 15;
  int kb   = khi * 2;                          // this lane's K sub-offset {0,2}
  const float* xr   = x + (size_t)(m0 + n) * CFEAT + kb;          // A row = m0 + (lane&15)
  const float* Wsel = W + ((n == 1) ? CFEAT : 0) + kb;            // n>=2 reads W harmlessly
  float mask = (n < 2) ? 1.f : 0.f;            // zero out B columns 2..15
  v8f acc = {0.f, 0.f, 0.f, 0.f, 0.f, 0.f, 0.f, 0.f};
  #pragma unroll
  for (int k0 = 0; k0 < CFEAT; k0 += 4) {
    v2f a  = *(const v2f*)(xr + k0);           // 8B-aligned
    v2f wv = *(const v2f*)(Wsel + k0);         // 8B-aligned, L0-resident after first tile
    v2f b  = { wv.x * mask, wv.y * mask };
    acc = __builtin_amdgcn_wmma_f32_16x16x4_f32(false, a, false, b, (short)0, acc, false, false);
  }
  if (n < 2) {                                 // only columns 0 (p0) and 1 (p1) are meaningful
    float* dstp = (n == 0) ? p0 : p1;
    int mbase = m0 + khi * 8;
    #pragma unroll
    for (int i = 0; i < 8; i++) dstp[mbase + i] = acc[i];
  }
}

// ---------------- K2: per-graph argmax of logit with min-index tie-break ----------------
__global__ void k_argmax(const float* __restrict__ p0, const float* __restrict__ p1,
                         const float* __restrict__ bptr, const int* __restrict__ src,
                         const int* __restrict__ dst, u64* __restrict__ table) {
  __shared__ u64 lt[NG];
  for (int i = threadIdx.x; i < NG; i += blockDim.x) lt[i] = 0ull;
  __syncthreads();
  float bb = bptr[0];
  int stride = gridDim.x * blockDim.x;
  for (int e = blockIdx.x * blockDim.x + threadIdx.x; e < NEDGE; e += stride) {
    int s = src[e], d = dst[e];
    float logit = p0[s] + p1[d] + bb;          // sigmoid is monotone -> same argmax
    unsigned u = __float_as_uint(logit);
    u ^= (u >> 31) ? 0xFFFFFFFFu : 0x80000000u;   // order-preserving map
    u64 key = ((u64)u << 32) | (unsigned)(~e);    // larger score, then smaller index wins
    atomicMax(&lt[s / NPG], key);
  }
  __syncthreads();
  for (int i = threadIdx.x; i < NG; i += blockDim.x) {
    u64 v = lt[i];
    if (v) atomicMax(&table[i], v);
  }
}

// ---------------- K3: decode winners ----------------
__global__ void k_winner(const u64* __restrict__ table, const int* __restrict__ src,
                         const int* __restrict__ dst, int* w0, int* w1) {
  int g = blockIdx.x * blockDim.x + threadIdx.x;
  if (g < NG) {
    unsigned e = ~(unsigned)(table[g] & 0xFFFFFFFFull);
    w0[g] = src[e]; w1[g] = dst[e];
  }
}

// ---------------- K4: new_x = [x[s0]+x[s1] ; x[survivors]] ----------------
__global__ void k_newx(const float* __restrict__ x, const int* __restrict__ w0,
                       const int* __restrict__ w1, float* __restrict__ outx) {
  int r = blockIdx.x, c = threadIdx.x;
  if (r < NG) {
    int s0 = w0[r], s1 = w1[r];
    outx[(size_t)r * CFEAT + c] = x[(size_t)s0 * CFEAT + c] + x[(size_t)s1 * CFEAT + c];
  } else {
    int t = r - NG, g = t / SURV, k = t % SURV;
    int l0 = w0[g] - g * NPG, l1 = w1[g] - g * NPG;
    int a = min(l0, l1), b = max(l0, l1);
    int l = k; if (l >= a) l++; if (l >= b) l++;   // k-th surviving local index
    outx[(size_t)r * CFEAT + c] = x[(size_t)(g * NPG + l) * CFEAT + c];
  }
}

// ---------------- cluster id in closed form (no global prefix sum needed) ----------------
__device__ __forceinline__ int cluster_of(int node, int g, int s0, int s1) {
  if (node == s0 || node == s1) return g;
  int l = node - g * NPG;
  int rank = l - (int)((s0 - g * NPG) < l) - (int)((s1 - g * NPG) < l);
  return NG + g * SURV + rank;
}

// ---------------- K5: scatter edge keys into per-graph bins ----------------
__global__ void k_scatter(const int* __restrict__ src, const int* __restrict__ dst,
                          const int* __restrict__ w0, const int* __restrict__ w1,
                          int* __restrict__ cursor, u64* __restrict__ grouped) {
  int stride = gridDim.x * blockDim.x;
  for (int e = blockIdx.x * blockDim.x + threadIdx.x; e < NEDGE; e += stride) {
    int s = src[e], d = dst[e];
    int g = s / NPG;                              // intra-graph edges: same g for d
    int s0 = w0[g], s1 = w1[g];
    int r = cluster_of(s, g, s0, s1);
    int c = cluster_of(d, g, s0, s1);
    u64 key = ((u64)(unsigned)r << 32) | (unsigned)c;   // u64 compare == lex (r,c)
    int pos = atomicAdd(&cursor[g], 1);
    if (pos < CAP) grouped[(size_t)g * CAP + pos] = key;
  }
}

// ---------------- K6: per-graph bitonic sort + dedup + compact; count r==g run ----------------
__global__ void __launch_bounds__(512)
k_sortgraph(u64* __restrict__ grouped, const int* __restrict__ cursor,
            int* __restrict__ u1, int* __restrict__ u2) {
  __shared__ u64 s[CAP];
  __shared__ int tsum[512];
  __shared__ int u1sh;
  int g = blockIdx.x, t = threadIdx.x;
  int cnt = cursor[g]; if (cnt > CAP) cnt = CAP;
  u64* gp = grouped + (size_t)g * CAP;
  for (int i = t; i < CAP; i += 512) s[i] = (i < cnt) ? gp[i] : 0xFFFFFFFFFFFFFFFFull;
  if (t == 0) u1sh = 0;
  for (int k = 2; k <= CAP; k <<= 1)
    for (int j = k >> 1; j > 0; j >>= 1) {
      __syncthreads();
      for (int i = t; i < CAP; i += 512) {
        int ixj = i ^ j;
        if (ixj > i) {
          u64 a = s[i], b = s[ixj];
          bool up = ((i & k) == 0);
          if (up ? (a > b) : (a < b)) { s[i] = b; s[ixj] = a; }
        }
      }
    }
  __syncthreads();
  // dedup flags over 8 consecutive elements per thread
  int base = t * 8, cntT = 0, flags = 0;
  for (int q = 0; q < 8; q++) {
    int i = base + q;
    u64 v = s[i];
    bool f = (v != 0xFFFFFFFFFFFFFFFFull) && (i == 0 || v != s[i - 1]);
    flags |= (int)f << q;
    cntT += f;
    if (f && (unsigned)(v >> 32) < (unsigned)NG) atomicAdd(&u1sh, 1);
  }
  tsum[t] = cntT;
  __syncthreads();
  for (int off = 1; off < 512; off <<= 1) {     // Hillis-Steele inclusive scan
    int v = (t >= off) ? tsum[t - off] : 0;
    __syncthreads();
    tsum[t] += v;
    __syncthreads();
  }
  int pos = tsum[t] - cntT;
  for (int q = 0; q < 8; q++)
    if ((flags >> q) & 1) gp[pos++] = s[base + q];  // compacted sorted-unique, in place
  if (t == 0) { u1[g] = u1sh; u2[g] = tsum[511] - u1sh; }
}

// ---------------- K7: global run offsets (two 512-wide scans) ----------------
__global__ void __launch_bounds__(512)
k_offsets(const int* __restrict__ u1, const int* __restrict__ u2,
          int* __restrict__ off1, int* __restrict__ off2) {
  __shared__ int a[NG], b[NG];
  int t = threadIdx.x;
  int va0 = u1[t], vb0 = u2[t];
  a[t] = va0; b[t] = vb0;
  __syncthreads();
  for (int off = 1; off < NG; off <<= 1) {
    int va = (t >= off) ? a[t - off] : 0;
    int vb = (t >= off) ? b[t - off] : 0;
    __syncthreads();
    a[t] += va; b[t] += vb;
    __syncthreads();
  }
  off1[t] = a[t] - va0;                  // exclusive prefix of run1 counts
  off2[t] = a[NG - 1] + b[t] - vb0;      // total1 + exclusive prefix of run2 counts
}

// ---------------- K8: emit coalesced edge list ----------------
__global__ void k_emit(const u64* __restrict__ grouped, const int* __restrict__ u1a,
                       const int* __restrict__ u2a, const int* __restrict__ off1,
                       const int* __restrict__ off2, int* __restrict__ outr,
                       int* __restrict__ outc) {
  int g = blockIdx.x;
  int n1 = u1a[g], u = n1 + u2a[g];
  int o1 = off1[g], o2 = off2[g];
  const u64* gp = grouped + (size_t)g * CAP;
  for (int i = threadIdx.x; i < u; i += blockDim.x) {
    u64 v = gp[i];
    int pos = (i < n1) ? o1 + i : o2 + (i - n1);
    outr[pos] = (int)(v >> 32);
    outc[pos] = (int)(v & 0xFFFFFFFFull);
  }
}

extern "C" void kernel_launch(void* const* d_in, const int* in_sizes, int n_in,
                              void* d_out, int out_size, void* d_ws, size_t ws_size,
                              hipStream_t stream) {
  const float* x    = (const float*)d_in[0];
  const int*   ei   = (const int*)d_in[1];
  const int*   src  = ei;
  const int*   dst  = ei + NEDGE;
  const float* W    = (const float*)d_in[3];
  const float* bias = (const float*)d_in[4];

  // output layout: new_x [NBNODES*CFEAT] f32 | new_edge_index [2*NEDGE] i32 | new_batch [NBNODES] i32
  float* out_x     = (float*)d_out;
  int*   out_ei    = (int*)d_out + (size_t)NBNODES * CFEAT;
  int*   out_r     = out_ei;
  int*   out_c     = out_ei + NEDGE;
  int*   out_batch = out_ei + 2 * NEDGE;

  // workspace layout (bytes)
  char* ws = (char*)d_ws;
  u64*  table   = (u64*)(ws + 0);                        //   4 KB
  float* p0     = (float*)(ws + 4096);                   // 390 KB
  float* p1     = (float*)(ws + 4096 + 4 * NNODES);
  int*  cursor  = (int*)(ws + 4096 + 8 * NNODES);
  int*  u1      = cursor + NG;
  int*  u2      = u1 + NG;
  int*  off1    = u2 + NG;
  int*  off2    = off1 + NG;
  int*  winS0   = off2 + NG;
  int*  winS1   = winS0 + NG;
  size_t goff   = 4096 + 8 * (size_t)NNODES + 8 * NG * sizeof(int);
  goff = (goff + 7) & ~(size_t)7;
  u64*  grouped = (u64*)(ws + goff);                     // 16 MB

  k_init<<<2048, 256, 0, stream>>>(out_ei, out_batch, table, cursor);
  k_project<<<(NNODES / 16) / 8, 256, 0, stream>>>(x, W, p0, p1);
  k_argmax<<<512, 256, 0, stream>>>(p0, p1, bias, src, dst, table);
  k_winner<<<2, 256, 0, stream>>>(table, src, dst, winS0, winS1);
  k_newx<<<NBNODES, CFEAT, 0, stream>>>(x, winS0, winS1, out_x);
  k_scatter<<<1024, 256, 0, stream>>>(src, dst, winS0, winS1, cursor, grouped);
  k_sortgraph<<<NG, 512, 0, stream>>>(grouped, cursor, u1, u2);
  k_offsets<<<1, NG, 0, stream>>>(u1, u2, off1, off2);
  k_emit<<<NG, 256, 0, stream>>>(grouped, u1, u2, off1, off2, out_r, out_c);
}